// RelationScorer_21148418966047
// MI455X (gfx1250) — compile-verified
//
#include <hip/hip_runtime.h>

typedef float v2f __attribute__((ext_vector_type(2)));
typedef float v8f __attribute__((ext_vector_type(8)));

#define BB 8
#define NN 256
#define DD 768
#define HH 768
#define CC 4
#define MS 103           // m = ceil(0.4*256)
#define MM (MS*MS)       // 10609

// CDNA5 dense fp32 WMMA: D = A(16x4) * B(4x16) + C(16x16)
__device__ __forceinline__ v8f wmma4(v2f a, v2f b, v8f c) {
  return __builtin_amdgcn_wmma_f32_16x16x4_f32(false, a, false, b, (short)0, c, false, false);
}

// ---------------------------------------------------------------------------
// Kernel 1: full span MLP. 128 rows per WG, fused 2-layer.
// logits[row][4], score[row] = mean(logits[1:4])
// ---------------------------------------------------------------------------
__global__ void __launch_bounds__(256)
span_mlp_kernel(const float* __restrict__ x, const float* __restrict__ W1,
                const float* __restrict__ b1, const float* __restrict__ W2,
                const float* __restrict__ b2, float* __restrict__ logits,
                float* __restrict__ score)
{
  __shared__ float At[128*34];         // x tile [row][d] stride 34 (8B-aligned frags)
  __shared__ float Bt[128*34];         // W1 tile transposed [h][d]
  __shared__ float wavePart[8*128*4];  // per-wave partial (row, c) sums

  const int tid  = threadIdx.x;
  const int wave = tid >> 5, lane = tid & 31, l16 = lane & 15;
  const int hi2  = (lane >> 4) * 2, hi8 = (lane >> 4) * 8;
  const int row0 = blockIdx.x * 128;

  for (int k = tid; k < 8*128*4; k += 256) wavePart[k] = 0.f;

  for (int hg = 0; hg < 6; ++hg) {
    const int hglob  = hg*128 + wave*16 + l16;     // this lane's hidden column
    const float bias = b1[hglob];
    const float4 w2v = *reinterpret_cast<const float4*>(W2 + (size_t)hglob*4);
    const v8f vz = {0.f,0.f,0.f,0.f,0.f,0.f,0.f,0.f};
    v8f acc[8];
#pragma unroll
    for (int mi = 0; mi < 8; ++mi) acc[mi] = vz;

    for (int kc = 0; kc < 24; ++kc) {
      const int d0 = kc*32;
      for (int idx = tid; idx < 4096; idx += 256) {
        int rl = idx >> 5, dl = idx & 31;
        At[rl*34 + dl] = x[(size_t)(row0 + rl)*DD + d0 + dl];
      }
      for (int idx = tid; idx < 4096; idx += 256) {
        int dl = idx >> 7, hl = idx & 127;
        Bt[hl*34 + dl] = W1[(size_t)(d0 + dl)*HH + hg*128 + hl];
      }
      __syncthreads();
#pragma unroll
      for (int ks = 0; ks < 8; ++ks) {
        const int dd = ks*4 + hi2;
        v2f bf = *reinterpret_cast<const v2f*>(&Bt[(wave*16 + l16)*34 + dd]);
#pragma unroll
        for (int mi = 0; mi < 8; ++mi) {
          v2f af = *reinterpret_cast<const v2f*>(&At[(mi*16 + l16)*34 + dd]);
          acc[mi] = wmma4(af, bf, acc[mi]);
        }
      }
      __syncthreads();
    }
    // epilogue: relu(z+b1) dot W2, reduce across the 16 lanes holding one row
#pragma unroll
    for (int mi = 0; mi < 8; ++mi) {
#pragma unroll
      for (int r = 0; r < 8; ++r) {
        float v = acc[mi][r] + bias;
        v = fmaxf(v, 0.f);
        float t0 = v*w2v.x, t1 = v*w2v.y, t2 = v*w2v.z, t3 = v*w2v.w;
#pragma unroll
        for (int off = 1; off < 16; off <<= 1) {
          t0 += __shfl_xor(t0, off, 32);
          t1 += __shfl_xor(t1, off, 32);
          t2 += __shfl_xor(t2, off, 32);
          t3 += __shfl_xor(t3, off, 32);
        }
        if (l16 == 0) {
          int p = mi*16 + r + hi8;
          float* wp = &wavePart[(wave*128 + p)*4];
          wp[0] += t0; wp[1] += t1; wp[2] += t2; wp[3] += t3;
        }
      }
    }
  }
  __syncthreads();
  if (tid < 128) {
    float s0 = b2[0], s1 = b2[1], s2 = b2[2], s3 = b2[3];
    for (int w = 0; w < 8; ++w) {                  // fixed order: deterministic
      const float* wp = &wavePart[(w*128 + tid)*4];
      s0 += wp[0]; s1 += wp[1]; s2 += wp[2]; s3 += wp[3];
    }
    int row = row0 + tid;
    float4 lg = {s0, s1, s2, s3};
    *reinterpret_cast<float4*>(logits + (size_t)row*4) = lg;
    score[row] = (s1 + s2 + s3) * (1.f/3.f);
  }
}

// ---------------------------------------------------------------------------
// Kernel 2: replicate the reference's selection exactly.
// order = stable argsort(-score); sel = ascending {k : order[k] < 103}
//       = ascending sorted {rank[i] : i < 103}, rank = stable descending rank.
// ---------------------------------------------------------------------------
__global__ void select_kernel(const float* __restrict__ score, int* __restrict__ sel)
{
  __shared__ float s[NN];
  __shared__ int flags[NN];
  const int b = blockIdx.x, tid = threadIdx.x;
  s[tid] = score[b*NN + tid];
  flags[tid] = 0;
  __syncthreads();
  const float si = s[tid];
  int rank = 0;
  for (int j = 0; j < NN; ++j) {
    float sj = s[j];
    rank += (sj > si) || (sj == si && j < tid);   // stable tie-break = jnp.argsort
  }
  if (tid < MS) flags[rank] = 1;                  // ranks are distinct: no race
  __syncthreads();
  if (tid == 0) {
    int c = 0;
    for (int k = 0; k < NN; ++k) if (flags[k]) sel[b*MS + (c++)] = k;
  }
}

// ---------------------------------------------------------------------------
// Kernel 3: gather x_ranked rows, selected logits, selected span ranges.
// ---------------------------------------------------------------------------
__global__ void gather_kernel(const float* __restrict__ x, const int* __restrict__ ranges,
                              const float* __restrict__ logits, const int* __restrict__ sel,
                              float* __restrict__ xr, float* __restrict__ lsr,
                              int* __restrict__ rr)
{
  const int bi = blockIdx.x;           // b*MS + i
  const int b = bi / MS;
  const int tid = threadIdx.x;
  const int src = sel[bi];
  const float* srow = x + ((size_t)b*NN + src)*DD;
  float* drow = xr + (size_t)bi*DD;
  for (int d = tid; d < DD; d += 256) drow[d] = srow[d];
  if (tid < 4) lsr[bi*4 + tid] = logits[((size_t)b*NN + src)*4 + tid];
  if (tid < 2) rr[bi*2 + tid] = ranges[src*2 + tid];
}

// ---------------------------------------------------------------------------
// Kernel 4: premix GEMM — preab[row, 0:768]  = x_r @ P1[0:768,:]
//                         preab[row, 768:1536]= x_r @ P1[768:1536,:]
// ---------------------------------------------------------------------------
__global__ void __launch_bounds__(256)
premix_kernel(const float* __restrict__ xr, const float* __restrict__ P1,
              float* __restrict__ preab)
{
  __shared__ float At[128*34];
  __shared__ float Bt[128*34];
  const int tid  = threadIdx.x;
  const int wave = tid >> 5, lane = tid & 31, l16 = lane & 15;
  const int hi2  = (lane >> 4) * 2, hi8 = (lane >> 4) * 8;
  const int mt = blockIdx.x / 12, nt = blockIdx.x % 12;
  const int row0 = mt*128, n0 = nt*128;
  const float* Wb = P1 + (n0 < HH ? (size_t)n0 : (size_t)HH*HH + (n0 - HH));

  const v8f vz = {0.f,0.f,0.f,0.f,0.f,0.f,0.f,0.f};
  v8f acc[8];
#pragma unroll
  for (int mi = 0; mi < 8; ++mi) acc[mi] = vz;

  for (int kc = 0; kc < 24; ++kc) {
    const int d0 = kc*32;
    for (int idx = tid; idx < 4096; idx += 256) {
      int rl = idx >> 5, dl = idx & 31;
      int row = min(row0 + rl, BB*MS - 1);
      At[rl*34 + dl] = xr[(size_t)row*DD + d0 + dl];
    }
    for (int idx = tid; idx < 4096; idx += 256) {
      int dl = idx >> 7, hl = idx & 127;
      Bt[hl*34 + dl] = Wb[(size_t)(d0 + dl)*HH + hl];
    }
    __syncthreads();
#pragma unroll
    for (int ks = 0; ks < 8; ++ks) {
      const int dd = ks*4 + hi2;
      v2f bf = *reinterpret_cast<const v2f*>(&Bt[(wave*16 + l16)*34 + dd]);
#pragma unroll
      for (int mi = 0; mi < 8; ++mi) {
        v2f af = *reinterpret_cast<const v2f*>(&At[(mi*16 + l16)*34 + dd]);
        acc[mi] = wmma4(af, bf, acc[mi]);
      }
    }
    __syncthreads();
  }
#pragma unroll
  for (int mi = 0; mi < 8; ++mi) {
#pragma unroll
    for (int r = 0; r < 8; ++r) {
      int p = mi*16 + r + hi8;
      int row = row0 + p;
      if (row < BB*MS)
        preab[(size_t)row*1536 + n0 + wave*16 + l16] = acc[mi][r];
    }
  }
}

// ---------------------------------------------------------------------------
// Kernel 5: the big fused pair kernel.
// Tile = 16 i's x 8 j's = 128 pairs. For all 768 hidden cols (6 groups x 128):
//   C = (x_i*x_j) @ P1c   via v_wmma_f32_16x16x4_f32
//   hp = relu(C + Apre_i + Bpre_j + bp1);  out4 += hp @ P2  (fused, per-wave)
// Final: out4 + bp2 + ls_i + ls_j, plus span_pair_ranges.
// ---------------------------------------------------------------------------
__global__ void __launch_bounds__(256)
pair_kernel(const float* __restrict__ xr, const float* __restrict__ preab,
            const float* __restrict__ P1, const float* __restrict__ bp1,
            const float* __restrict__ P2, const float* __restrict__ bp2,
            const float* __restrict__ lsr, const int* __restrict__ rr,
            float* __restrict__ out_sum, int* __restrict__ out_rng)
{
  extern __shared__ float smem[];
  float* xi_s     = smem;               // 16*768
  float* xj_s     = xi_s + 16*768;      //  8*768
  float* Zt       = xj_s + 8*768;       // 128*34
  float* Bt       = Zt + 128*34;        // 128*34
  float* As       = Bt + 128*34;        // 16*128  (Apre tile)
  float* Bs       = As + 16*128;        //  8*128  (Bpre tile)
  float* wavePart = Bs + 8*128;         //  8*128*4

  const int tid  = threadIdx.x;
  const int wave = tid >> 5, lane = tid & 31, l16 = lane & 15;
  const int hi2  = (lane >> 4) * 2, hi8 = (lane >> 4) * 8;
  const int bid = blockIdx.x;
  const int b = bid / 91, r91 = bid % 91;
  const int it = r91 / 13, jt = r91 % 13;
  const int i0 = it*16, j0 = jt*8;

  for (int k = tid; k < 8*128*4; k += 256) wavePart[k] = 0.f;
  // stage the 16 x-rows (i) and 8 x-rows (j) for this tile (reused 144x)
  for (int idx = tid; idx < 16*768; idx += 256) {
    int il = idx / 768, d = idx - il*768;
    int row = b*MS + min(i0 + il, MS - 1);
    xi_s[idx] = xr[(size_t)row*DD + d];
  }
  for (int idx = tid; idx < 8*768; idx += 256) {
    int jl = idx / 768, d = idx - jl*768;
    int row = b*MS + min(j0 + jl, MS - 1);
    xj_s[idx] = xr[(size_t)row*DD + d];
  }
  __syncthreads();

  const float* P1c = P1 + (size_t)2*HH*HH;   // P1 rows 1536:2304

  for (int hg = 0; hg < 6; ++hg) {
    for (int idx = tid; idx < 16*128; idx += 256) {
      int il = idx >> 7, hl = idx & 127;
      int row = b*MS + min(i0 + il, MS - 1);
      As[idx] = preab[(size_t)row*1536 + hg*128 + hl];
    }
    for (int idx = tid; idx < 8*128; idx += 256) {
      int jl = idx >> 7, hl = idx & 127;
      int row = b*MS + min(j0 + jl, MS - 1);
      Bs[idx] = preab[(size_t)row*1536 + 768 + hg*128 + hl];
    }
    const int hglob  = hg*128 + wave*16 + l16;
    const float bias = bp1[hglob];
    const float4 p2v = *reinterpret_cast<const float4*>(P2 + (size_t)hglob*4);
    const v8f vz = {0.f,0.f,0.f,0.f,0.f,0.f,0.f,0.f};
    v8f acc[8];
#pragma unroll
    for (int mi = 0; mi < 8; ++mi) acc[mi] = vz;

    for (int kc = 0; kc < 24; ++kc) {
      const int d0 = kc*32;
      // build Z tile: Z[p][dl] = xi[p>>3][d]*xj[p&7][d]
      for (int idx = tid; idx < 4096; idx += 256) {
        int p = idx >> 5, dl = idx & 31;
        Zt[p*34 + dl] = xi_s[(p>>3)*768 + d0 + dl] * xj_s[(p&7)*768 + d0 + dl];
      }
      // stage P1c chunk transposed: Bt[h][d]
      for (int idx = tid; idx < 4096; idx += 256) {
        int dl = idx >> 7, hl = idx & 127;
        Bt[hl*34 + dl] = P1c[(size_t)(d0 + dl)*HH + hg*128 + hl];
      }
      __syncthreads();
#pragma unroll
      for (int ks = 0; ks < 8; ++ks) {
        const int dd = ks*4 + hi2;
        v2f bf = *reinterpret_cast<const v2f*>(&Bt[(wave*16 + l16)*34 + dd]);
#pragma unroll
        for (int mi = 0; mi < 8; ++mi) {
          v2f af = *reinterpret_cast<const v2f*>(&Zt[(mi*16 + l16)*34 + dd]);
          acc[mi] = wmma4(af, bf, acc[mi]);
        }
      }
      __syncthreads();
    }
    // epilogue: relu(C + Apre + Bpre + bp1) dot P2, butterfly-reduce 16 lanes
#pragma unroll
    for (int mi = 0; mi < 8; ++mi) {
#pragma unroll
      for (int r = 0; r < 8; ++r) {
        int p = mi*16 + r + hi8;
        float v = acc[mi][r]
                + As[(p>>3)*128 + wave*16 + l16]
                + Bs[(p&7)*128 + wave*16 + l16] + bias;
        v = fmaxf(v, 0.f);
        float t0 = v*p2v.x, t1 = v*p2v.y, t2 = v*p2v.z, t3 = v*p2v.w;
#pragma unroll
        for (int off = 1; off < 16; off <<= 1) {
          t0 += __shfl_xor(t0, off, 32);
          t1 += __shfl_xor(t1, off, 32);
          t2 += __shfl_xor(t2, off, 32);
          t3 += __shfl_xor(t3, off, 32);
        }
        if (l16 == 0) {
          float* wp = &wavePart[(wave*128 + p)*4];
          wp[0] += t0; wp[1] += t1; wp[2] += t2; wp[3] += t3;
        }
      }
    }
    __syncthreads();   // protect As/Bs before next h-group restage
  }

  if (tid < 128) {
    const int p = tid, il = p >> 3, jl = p & 7;
    const int i = i0 + il, j = j0 + jl;
    if (i < MS && j < MS) {
      float s0 = bp2[0], s1 = bp2[1], s2 = bp2[2], s3 = bp2[3];
      for (int w = 0; w < 8; ++w) {            // fixed order: deterministic
        const float* wp = &wavePart[(w*128 + p)*4];
        s0 += wp[0]; s1 += wp[1]; s2 += wp[2]; s3 += wp[3];
      }
      const float* li = lsr + ((size_t)b*MS + i)*4;
      const float* lj = lsr + ((size_t)b*MS + j)*4;
      const size_t gp = (size_t)b*MM + (size_t)i*MS + j;
      float4 o = { s0 + li[0] + lj[0], s1 + li[1] + lj[1],
                   s2 + li[2] + lj[2], s3 + li[3] + lj[3] };
      *reinterpret_cast<float4*>(out_sum + gp*4) = o;
      const int* ri = rr + ((size_t)b*MS + i)*2;
      const int* rj = rr + ((size_t)b*MS + j)*2;
      int4 rg = { ri[0], ri[1], rj[0], rj[1] };
      *reinterpret_cast<int4*>(out_rng + gp*4) = rg;
    }
  }
}

// ---------------------------------------------------------------------------
extern "C" void kernel_launch(void* const* d_in, const int* in_sizes, int n_in,
                              void* d_out, int out_size, void* d_ws, size_t ws_size,
                              hipStream_t stream) {
  (void)in_sizes; (void)n_in; (void)out_size; (void)ws_size;
  const float* x   = (const float*)d_in[0];
  const int*   sr  = (const int*)  d_in[1];
  const float* W1  = (const float*)d_in[2];
  const float* b1  = (const float*)d_in[3];
  const float* W2  = (const float*)d_in[4];
  const float* b2  = (const float*)d_in[5];
  const float* P1  = (const float*)d_in[6];
  const float* bp1 = (const float*)d_in[7];
  const float* P2  = (const float*)d_in[8];
  const float* bp2 = (const float*)d_in[9];

  float* ws     = (float*)d_ws;
  float* logits = ws;                    // 8192 f
  float* score  = ws + 8192;             // 2048 f
  int*   sel    = (int*)(ws + 10240);    // 824 i
  float* lsr    = ws + 11072;            // 3296 f
  int*   rngr   = (int*)(ws + 14368);    // 1648 i
  float* xrk    = ws + 16128;            // 8*103*768 f
  float* preab  = ws + 648960;           // 824*1536 f   (end ~7.7 MB)

  float* out_sum = (float*)d_out;
  int*   out_rng = (int*)d_out + (size_t)BB*MM*CC;

  span_mlp_kernel<<<16, 256, 0, stream>>>(x, W1, b1, W2, b2, logits, score);
  select_kernel<<<BB, 256, 0, stream>>>(score, sel);
  gather_kernel<<<BB*MS, 256, 0, stream>>>(x, sr, logits, sel, xrk, lsr, rngr);
  premix_kernel<<<7*12, 256, 0, stream>>>(xrk, P1, preab);
  const size_t smem = (size_t)(16*768 + 8*768 + 2*128*34 + 16*128 + 8*128 + 8*128*4)
                      * sizeof(float);   // ~134 KB of the 320 KB WGP LDS
  pair_kernel<<<BB*91, 256, smem, stream>>>(xrk, preab, P1, bp1, P2, bp2,
                                            lsr, rngr, out_sum, out_rng);
}